// SpatioTemporalHawkes_89352499626690
// MI455X (gfx1250) — compile-verified
//
#include <hip/hip_runtime.h>
#include <math.h>

typedef __attribute__((ext_vector_type(2))) float v2f;
typedef __attribute__((ext_vector_type(8))) float v8f;

#define BB 4
#define NN 4096
#define NT (NN / 16)          // 256 row-tiles per batch
#define NTP (NT / 2)          // 128 paired row-tile jobs per batch
#define WAVES_PER_BLOCK 8

// One wave handles TWO 16-row tiles of one batch: tiles g and NT-1-g.
// Every wave sweeps exactly NT+1 column tiles -> perfect load balance; the
// B fragment is shared between the two WMMAs while both tiles are active.
// The A fragment is pre-scaled by log2(e) so the WMMA emits the exponent in
// base-2 domain and v_exp_f32 consumes it directly (no pk_mul conversion).
__global__ __launch_bounds__(256) void hawkes_tiles(
    const float* __restrict__ loc,          // [B,N,2]
    const float* __restrict__ times,        // [B,N] sorted ascending per batch
    const unsigned char* __restrict__ mask, // [B,N] bool (1 byte)
    const float* __restrict__ mu_p,
    const float* __restrict__ alpha_p,
    const float* __restrict__ beta_p,
    const float* __restrict__ sigma_p,
    const float* __restrict__ slow,         // [2]
    const float* __restrict__ shigh,        // [2]
    const float* __restrict__ tend_p,
    float* __restrict__ partial)            // [B*NTP] per-wave partial results
{
  const int lane = threadIdx.x & 31;
  const int wib  = threadIdx.x >> 5;
  // Wave-uniform scalar job indices -> scalar loop control, no exec masking.
  const int gw = __builtin_amdgcn_readfirstlane(blockIdx.x * WAVES_PER_BLOCK + wib);
  const int b  = __builtin_amdgcn_readfirstlane(gw / NTP);
  const int g  = __builtin_amdgcn_readfirstlane(gw % NTP);
  const int it0 = g;            // small tile (g+1 column tiles)
  const int it1 = NT - 1 - g;   // large tile (NT-g column tiles)

  const float mu    = mu_p[0];
  const float alpha = alpha_p[0];
  const float beta  = beta_p[0];
  const float sigma = sigma_p[0];
  const float t_end = tend_p[0];

  const float s2     = sigma * sigma;
  const float invs2  = 1.0f / s2;
  const float inv2s2 = 0.5f * invs2;
  const float norm   = beta / (2.0f * 3.14159265358979323846f * s2); // d=2
  const float L2E    = 1.44269504088896340736f;  // log2(e), folded into A

  const int half = lane >> 4;   // K-half for A/B fragments
  const int sub  = lane & 15;   // row (A) / col (B,C,D) within tile
  const int base = b * NN;

  // ---- A fragments (pre-scaled by log2e): 16x4 f32.
  //      lanes 0-15: K=0,1 ; lanes 16-31: K=2,3 ----
  v2f afrag0, afrag1;
  {
    const int gi0 = it0 * 16 + sub;
    const float x = loc[(base + gi0) * 2 + 0];
    const float y = loc[(base + gi0) * 2 + 1];
    const float t = times[base + gi0];
    if (half == 0) { afrag0.x = x * invs2 * L2E; afrag0.y = y * invs2 * L2E; }
    else { afrag0.x = L2E; afrag0.y = (-beta * t - (x * x + y * y) * inv2s2) * L2E; }
  }
  {
    const int gi1 = it1 * 16 + sub;
    const float x = loc[(base + gi1) * 2 + 0];
    const float y = loc[(base + gi1) * 2 + 1];
    const float t = times[base + gi1];
    if (half == 0) { afrag1.x = x * invs2 * L2E; afrag1.y = y * invs2 * L2E; }
    else { afrag1.x = L2E; afrag1.y = (-beta * t - (x * x + y * y) * inv2s2) * L2E; }
  }

  // Row times for the causal mask: C/D row m = v + 8*half
  float tr0[8], tr1[8];
#pragma unroll
  for (int v = 0; v < 8; ++v) {
    tr0[v] = times[base + it0 * 16 + 8 * half + v];
    tr1[v] = times[base + it1 * 16 + 8 * half + v];
  }

  float rs0[8], rs1[8];
#pragma unroll
  for (int v = 0; v < 8; ++v) { rs0[v] = 0.0f; rs1[v] = 0.0f; }

  // ---- sweep column tiles; both row tiles share each B fragment ----
  for (int jt = 0; jt <= it1; ++jt) {
    const int gj   = jt * 16 + sub;
    const float xj = loc[(base + gj) * 2 + 0];
    const float yj = loc[(base + gj) * 2 + 1];
    const float tj = times[base + gj];
    const bool  mj = (mask[base + gj] != 0);

    // B fragment: 4x16 f32. lanes 0-15: K=0,1 ; lanes 16-31: K=2,3
    v2f bfrag;
    if (half == 0) { bfrag.x = xj; bfrag.y = yj; }
    else { bfrag.x = beta * tj - (xj * xj + yj * yj) * inv2s2; bfrag.y = 1.0f; }

    v8f zero = {};
    {
      // e[v] = log2e * (-beta*dt - sqdist/(2 sigma^2))  -> feed v_exp_f32 directly
      v8f e = __builtin_amdgcn_wmma_f32_16x16x4_f32(
          false, afrag1, false, bfrag, (short)0, zero, false, false);
#pragma unroll
      for (int v = 0; v < 8; ++v) {
        const bool valid = mj && (tr1[v] > tj);
        rs1[v] += valid ? __builtin_amdgcn_exp2f(e[v]) : 0.0f;
      }
    }
    if (jt <= it0) {   // uniform (scalar) branch
      v8f e = __builtin_amdgcn_wmma_f32_16x16x4_f32(
          false, afrag0, false, bfrag, (short)0, zero, false, false);
#pragma unroll
      for (int v = 0; v < 8; ++v) {
        const bool valid = mj && (tr0[v] > tj);
        rs0[v] += valid ? __builtin_amdgcn_exp2f(e[v]) : 0.0f;
      }
    }
  }

  // ---- reduce row sums across the 16 lanes of each half ----
#pragma unroll
  for (int v = 0; v < 8; ++v) {
    float s0 = rs0[v], s1 = rs1[v];
    s0 += __shfl_xor(s0, 1);  s1 += __shfl_xor(s1, 1);
    s0 += __shfl_xor(s0, 2);  s1 += __shfl_xor(s1, 2);
    s0 += __shfl_xor(s0, 4);  s1 += __shfl_xor(s1, 4);
    s0 += __shfl_xor(s0, 8);  s1 += __shfl_xor(s1, 8);
    rs0[v] = s0;  rs1[v] = s1;  // full row sums for rows 8*half+v of each tile
  }

  // ---- per-event log-intensity + compensator for this wave's 32 rows ----
  float contrib = 0.0f;
  if (sub == 0) {  // lanes 0 (rows 0..7) and 16 (rows 8..15) of each tile
    const float inv_ssqrt2 = 1.0f / (sigma * 1.41421356237309504880f);
    const float lx = slow[0],  ly = slow[1];
    const float hx = shigh[0], hy = shigh[1];
#pragma unroll
    for (int tile = 0; tile < 2; ++tile) {
      const int   ib = (tile == 0 ? it0 : it1) * 16;
      const float* rsp = (tile == 0) ? rs0 : rs1;
      const float* trp = (tile == 0) ? tr0 : tr1;
#pragma unroll
      for (int v = 0; v < 8; ++v) {
        const int r = ib + 8 * half + v;
        if (mask[base + r] != 0) {
          const float lam = mu + alpha * norm * rsp[v];
          contrib += __logf(lam);
          const float x  = loc[(base + r) * 2 + 0];
          const float y  = loc[(base + r) * 2 + 1];
          const float cx = 0.5f * (erff((hx - x) * inv_ssqrt2) - erff((lx - x) * inv_ssqrt2));
          const float cy = 0.5f * (erff((hy - y) * inv_ssqrt2) - erff((ly - y) * inv_ssqrt2));
          const float tm = 1.0f - __expf(-beta * (t_end - trp[v]));
          contrib -= alpha * tm * cx * cy;
        }
      }
    }
  }
  contrib += __shfl_xor(contrib, 16);
  if (lane == 0) partial[gw] = contrib;
}

// Fixed-order deterministic reduction of per-wave partials into out[b].
__global__ void hawkes_reduce(const float* __restrict__ partial,
                              const float* __restrict__ mu_p,
                              const float* __restrict__ slow,
                              const float* __restrict__ shigh,
                              const float* __restrict__ tend_p,
                              float* __restrict__ out)
{
  const int b = threadIdx.x;
  if (b < BB) {
    const float area = (shigh[0] - slow[0]) * (shigh[1] - slow[1]);
    float s = -mu_p[0] * area * tend_p[0];   // constant compensator term
    for (int i = 0; i < NTP; ++i) s += partial[b * NTP + i];
    out[b] = s;
  }
}

extern "C" void kernel_launch(void* const* d_in, const int* in_sizes, int n_in,
                              void* d_out, int out_size, void* d_ws, size_t ws_size,
                              hipStream_t stream) {
  const float*         loc     = (const float*)d_in[0];         // [B,N,2]
  const float*         times   = (const float*)d_in[1];         // [B,N]
  const unsigned char* mask    = (const unsigned char*)d_in[2]; // [B,N] bool
  const float*         mu      = (const float*)d_in[3];
  const float*         alpha   = (const float*)d_in[4];
  const float*         beta    = (const float*)d_in[5];
  const float*         sigma   = (const float*)d_in[6];
  const float*         slow    = (const float*)d_in[7];         // [2]
  const float*         shigh   = (const float*)d_in[8];         // [2]
  const float*         t_end   = (const float*)d_in[9];
  float*               out     = (float*)d_out;                 // [B]
  float*               partial = (float*)d_ws;                  // B*NTP floats = 2 KB

  const int total_waves = BB * NTP;                      // 512
  const int blocks = total_waves / WAVES_PER_BLOCK;      // 64 blocks x 256 threads
  hawkes_tiles<<<blocks, 32 * WAVES_PER_BLOCK, 0, stream>>>(
      loc, times, mask, mu, alpha, beta, sigma, slow, shigh, t_end, partial);
  hawkes_reduce<<<1, 32, 0, stream>>>(partial, mu, slow, shigh, t_end, out);
}